// WassDistLoss_25632364822595
// MI455X (gfx1250) — compile-verified
//
#include <hip/hip_runtime.h>
#include <math.h>

typedef float v2f __attribute__((ext_vector_type(2)));
typedef float v8f __attribute__((ext_vector_type(8)));

#define B_      16
#define N_      1024
#define M_      1024
#define D_      32
#define EPS_    1e-3f
#define INV_EPS 1000.0f
#define THRESH_ 0.1f
#define LOGMU_  (-6.9314616f)   // log(1/1024 + 1e-8)

// ---------------- init: zero u, v, done flag, output ----------------
__global__ void sink_init(float* __restrict__ u, float* __restrict__ v,
                          int* __restrict__ done, float* __restrict__ out) {
    int idx = blockIdx.x * 256 + threadIdx.x;
    if (idx < B_ * N_) { u[idx] = 0.0f; v[idx] = 0.0f; }
    if (idx == 0) { *done = 0; *out = 0.0f; }
}

// ---------------- squared row norms of X and Y ----------------
__global__ void sink_norms(const float* __restrict__ X, const float* __restrict__ Y,
                           float* __restrict__ xn, float* __restrict__ yn) {
    int g = blockIdx.x * 256 + threadIdx.x;          // 32768 rows total
    const float* src = (g < B_ * N_) ? X : Y;
    int r = (g < B_ * N_) ? g : g - B_ * N_;
    const float4* row4 = (const float4*)(src + (size_t)r * D_);
    float s = 0.0f;
#pragma unroll
    for (int d = 0; d < D_ / 4; ++d) {
        float4 x = row4[d];
        s += x.x * x.x + x.y * x.y + x.z * x.z + x.w * x.w;
    }
    if (g < B_ * N_) xn[r] = s; else yn[r] = s;
}

// ---------------- C = xn + yn - 2*X@Y^T via V_WMMA_F32_16X16X4_F32 ----------------
// one wave per 16x16 tile; 8 k-steps of K=4 over D=32
__global__ void __launch_bounds__(256) sink_cost(const float* __restrict__ X,
                                                 const float* __restrict__ Y,
                                                 const float* __restrict__ xn,
                                                 const float* __restrict__ yn,
                                                 float* __restrict__ C) {
    int wave = blockIdx.x * 8 + (threadIdx.x >> 5);
    int lane = threadIdx.x & 31;
    int b    = wave >> 12;            // 64*64 = 4096 tiles per batch
    int rest = wave & 4095;
    int i0   = (rest >> 6) << 4;
    int j0   = (rest & 63) << 4;
    int hi   = lane >> 4;             // half-wave select
    int lo   = lane & 15;

    const float* Xb = X + ((size_t)b * N_) * D_;
    const float* Yb = Y + ((size_t)b * M_) * D_;

    v8f acc = {0.f, 0.f, 0.f, 0.f, 0.f, 0.f, 0.f, 0.f};
#pragma unroll
    for (int k = 0; k < D_; k += 4) {
        int kk = k + 2 * hi;
        v2f a, bmat;
        // A: 16x4 f32 tile — lanes 0-15: rows, VGPR0=K=k,VGPR1=K=k+1; lanes 16-31: K=k+2,k+3
        a.x    = Xb[(size_t)(i0 + lo) * D_ + kk];
        a.y    = Xb[(size_t)(i0 + lo) * D_ + kk + 1];
        // B: 4x16 f32 tile — B[kk][n] = Y[j0+n][kk] (mirrored layout)
        bmat.x = Yb[(size_t)(j0 + lo) * D_ + kk];
        bmat.y = Yb[(size_t)(j0 + lo) * D_ + kk + 1];
        acc = __builtin_amdgcn_wmma_f32_16x16x4_f32(false, a, false, bmat,
                                                    (short)0, acc, false, false);
    }

    size_t base = ((size_t)b << 20);
    float  ync  = yn[((size_t)b << 10) + j0 + lo];
#pragma unroll
    for (int r = 0; r < 8; ++r) {
        int row = i0 + r + 8 * hi;    // C/D layout: VGPR r -> rows r / r+8
        float cij = xn[((size_t)b << 10) + row] + ync - 2.0f * acc[r];
        C[base + ((size_t)row << 10) + (j0 + lo)] = cij;
    }
}

__device__ __forceinline__ void lse_merge(float& m, float& s, float mo, float so) {
    float nm = fmaxf(m, mo);
    s = s * __expf(m - nm) + so * __expf(mo - nm);
    m = nm;
}

// ---------------- row logsumexp: u_new = eps*(logmu - LSE_j((v_j - C_ij)/eps)) --------
// one wave per row; float4 streaming (512B per wave per step); 4 parallel accumulators
__global__ void __launch_bounds__(256) sink_row(const float* __restrict__ C,
                                                float* __restrict__ u,
                                                const float* __restrict__ v,
                                                float* __restrict__ partial,
                                                const int* __restrict__ done) {
    if (*done) return;
    __shared__ float wdu[8];
    int widx = threadIdx.x >> 5, lane = threadIdx.x & 31;
    int w = blockIdx.x * 8 + widx;
    int b = w >> 10, i = w & 1023;
    const float4* C4 = (const float4*)(C + ((size_t)b << 20) + ((size_t)i << 10));
    const float4* v4 = (const float4*)(v + ((size_t)b << 10));

    float m0 = -__builtin_inff(), m1 = m0, m2 = m0, m3 = m0;
    float s0 = 0.f, s1 = 0.f, s2 = 0.f, s3 = 0.f;
#pragma unroll
    for (int t = 0; t < 8; ++t) {
        int j4 = (t << 5) + lane;                     // coalesced float4
        float4 c = C4[j4];
        float4 vv = v4[j4];
        float z0 = (vv.x - c.x) * INV_EPS;
        float z1 = (vv.y - c.y) * INV_EPS;
        float z2 = (vv.z - c.z) * INV_EPS;
        float z3 = (vv.w - c.w) * INV_EPS;
        float n0 = fmaxf(m0, z0); s0 = s0 * __expf(m0 - n0) + __expf(z0 - n0); m0 = n0;
        float n1 = fmaxf(m1, z1); s1 = s1 * __expf(m1 - n1) + __expf(z1 - n1); m1 = n1;
        float n2 = fmaxf(m2, z2); s2 = s2 * __expf(m2 - n2) + __expf(z2 - n2); m2 = n2;
        float n3 = fmaxf(m3, z3); s3 = s3 * __expf(m3 - n3) + __expf(z3 - n3); m3 = n3;
    }
    lse_merge(m0, s0, m1, s1);
    lse_merge(m2, s2, m3, s3);
    lse_merge(m0, s0, m2, s2);
    // wave32 (m,s) reduction
    for (int off = 16; off >= 1; off >>= 1) {
        float mo = __shfl_xor(m0, off, 32);
        float so = __shfl_xor(s0, off, 32);
        lse_merge(m0, s0, mo, so);
    }
    if (lane == 0) {
        float L  = m0 + __logf(s0);
        float un = EPS_ * (LOGMU_ - L);
        size_t idx = ((size_t)b << 10) + i;
        wdu[widx] = fabsf(un - u[idx]);
        u[idx] = un;
    }
    __syncthreads();
    if (threadIdx.x == 0) {
        float t = 0.0f;
#pragma unroll
        for (int k = 0; k < 8; ++k) t += wdu[k];
        partial[blockIdx.x] = t;
    }
}

// ---------------- col logsumexp: v_new = eps*(lognu - LSE_i((u_i - C_ij)/eps)) --------
// block: (b, 64-col chunk); 16 i-slices of 64 rows; each thread streams float4 (4 cols)
__global__ void __launch_bounds__(256) sink_col(const float* __restrict__ C,
                                                const float* __restrict__ u,
                                                float* __restrict__ v,
                                                const int* __restrict__ done) {
    if (*done) return;
    __shared__ float sm[256 * 4], ss[256 * 4];
    int b     = blockIdx.x >> 4;
    int chunk = blockIdx.x & 15;
    int t   = threadIdx.x;
    int jl4 = t & 15;            // which float4 group within 64-col chunk
    int isl = t >> 4;            // i-slice: 16 slices x 64 rows
    const float* ub = u + ((size_t)b << 10);
    const float4* C4 = (const float4*)(C + ((size_t)b << 20) + (chunk << 6)) + jl4;

    float m0 = -__builtin_inff(), m1 = m0, m2 = m0, m3 = m0;
    float s0 = 0.f, s1 = 0.f, s2 = 0.f, s3 = 0.f;
    int ibase = isl << 6;
#pragma unroll 4
    for (int ii = 0; ii < 64; ++ii) {
        int i = ibase + ii;
        float4 c = C4[(size_t)i << 8];               // row stride = 256 float4
        float ui = ub[i];                            // broadcast
        float z0 = (ui - c.x) * INV_EPS;
        float z1 = (ui - c.y) * INV_EPS;
        float z2 = (ui - c.z) * INV_EPS;
        float z3 = (ui - c.w) * INV_EPS;
        float n0 = fmaxf(m0, z0); s0 = s0 * __expf(m0 - n0) + __expf(z0 - n0); m0 = n0;
        float n1 = fmaxf(m1, z1); s1 = s1 * __expf(m1 - n1) + __expf(z1 - n1); m1 = n1;
        float n2 = fmaxf(m2, z2); s2 = s2 * __expf(m2 - n2) + __expf(z2 - n2); m2 = n2;
        float n3 = fmaxf(m3, z3); s3 = s3 * __expf(m3 - n3) + __expf(z3 - n3); m3 = n3;
    }
    int pb = t << 2;
    sm[pb + 0] = m0; ss[pb + 0] = s0;
    sm[pb + 1] = m1; ss[pb + 1] = s1;
    sm[pb + 2] = m2; ss[pb + 2] = s2;
    sm[pb + 3] = m3; ss[pb + 3] = s3;
    __syncthreads();
    if (t < 64) {
        // thread t owns column (chunk*64 + t): float4-group t>>2, component t&3
        int idx0 = ((t >> 2) << 2) + (t & 3);        // slice-0 partial index
        float M = sm[idx0], S = ss[idx0];
#pragma unroll
        for (int k = 1; k < 16; ++k)
            lse_merge(M, S, sm[(k << 6) + idx0], ss[(k << 6) + idx0]);
        float L = M + __logf(S);
        v[((size_t)b << 10) + (chunk << 6) + t] = EPS_ * (LOGMU_ - L);
    }
}

// ---------------- convergence check: err = (sum partials)/B ----------------
__global__ void sink_check(const float* __restrict__ partial, int* __restrict__ done) {
    if (*done) return;
    __shared__ float red[256];
    float s = 0.0f;
    for (int k = threadIdx.x; k < 2048; k += 256) s += partial[k];
    red[threadIdx.x] = s;
    __syncthreads();
    for (int off = 128; off >= 1; off >>= 1) {
        if (threadIdx.x < off) red[threadIdx.x] += red[threadIdx.x + off];
        __syncthreads();
    }
    if (threadIdx.x == 0 && (red[0] * (1.0f / (float)B_)) < THRESH_) *done = 1;
}

// ---------------- final: out = mean_b sum_ij exp((u+v-C)/eps)*C ----------------
__global__ void __launch_bounds__(256) sink_final(const float* __restrict__ C,
                                                  const float* __restrict__ u,
                                                  const float* __restrict__ v,
                                                  float* __restrict__ out) {
    __shared__ float red[256];
    int b     = blockIdx.x >> 6;
    int chunk = blockIdx.x & 63;          // 16 rows per block
    const float*  ub = u + ((size_t)b << 10);
    const float4* v4 = (const float4*)(v + ((size_t)b << 10));
    const float4* C4 = (const float4*)(C + ((size_t)b << 20) + ((size_t)(chunk << 4) << 10));

    float local = 0.0f;
    for (int e = threadIdx.x; e < 16 * 256; e += 256) {   // float4 elements
        int r  = e >> 8;                 // row within chunk
        int j4 = e & 255;
        float4 c  = C4[(r << 8) + j4];
        float4 vv = v4[j4];
        float ui  = ub[(chunk << 4) + r];
        local += __expf((ui + vv.x - c.x) * INV_EPS) * c.x;
        local += __expf((ui + vv.y - c.y) * INV_EPS) * c.y;
        local += __expf((ui + vv.z - c.z) * INV_EPS) * c.z;
        local += __expf((ui + vv.w - c.w) * INV_EPS) * c.w;
    }
    red[threadIdx.x] = local;
    __syncthreads();
    for (int off = 128; off >= 1; off >>= 1) {
        if (threadIdx.x < off) red[threadIdx.x] += red[threadIdx.x + off];
        __syncthreads();
    }
    if (threadIdx.x == 0) atomicAdd(out, red[0] * (1.0f / (float)B_));
}

extern "C" void kernel_launch(void* const* d_in, const int* in_sizes, int n_in,
                              void* d_out, int out_size, void* d_ws, size_t ws_size,
                              hipStream_t stream) {
    const float* X = (const float*)d_in[0];   // [16,1024,32]
    const float* Y = (const float*)d_in[1];   // [16,1024,32]
    float* out = (float*)d_out;               // scalar

    char*  ws      = (char*)d_ws;
    float* C       = (float*)ws;                                  // 64 MB
    float* xn      = (float*)(ws + (size_t)B_ * N_ * M_ * 4);     // 16384
    float* yn      = xn + B_ * N_;                                // 16384
    float* u       = yn + B_ * M_;                                // 16384
    float* v       = u + B_ * N_;                                 // 16384
    float* partial = v + B_ * M_;                                 // 2048
    int*   done    = (int*)(partial + 2048);

    sink_init <<<64,   256, 0, stream>>>(u, v, done, out);
    sink_norms<<<128,  256, 0, stream>>>(X, Y, xn, yn);
    sink_cost <<<8192, 256, 0, stream>>>(X, Y, xn, yn, C);

    for (int it = 0; it < 100; ++it) {
        sink_row  <<<2048, 256, 0, stream>>>(C, u, v, partial, done);
        sink_col  <<<256,  256, 0, stream>>>(C, u, v, done);
        sink_check<<<1,    256, 0, stream>>>(partial, done);
    }

    sink_final<<<1024, 256, 0, stream>>>(C, u, v, out);
}